// MaskChamferDistance_14121852469352
// MI455X (gfx1250) — compile-verified
//
#include <hip/hip_runtime.h>
#include <hip/hip_bf16.h>

typedef __attribute__((ext_vector_type(2)))  float    v2f;
typedef __attribute__((ext_vector_type(8)))  float    v8f;

static constexpr int B_  = 8;
static constexpr int N_  = 2048;
static constexpr int M_  = 2048;
static constexpr int NT  = N_ / 16;     // 128 source tiles per batch
static constexpr int MT  = M_ / 16;     // 128 target tiles per batch
static constexpr int MCH = 8;           // m-chunks per (b, source tile) -> 8192 waves
static constexpr int MTC = MT / MCH;    // 16 target tiles per wave

__device__ __forceinline__ float finf() { return __builtin_inff(); }

// ---------------------------------------------------------------------------
// Init workspace: all three min arrays = +inf (ordered-uint encoding),
// maxVal = 0 (distances are nonnegative).
// ---------------------------------------------------------------------------
__global__ __launch_bounds__(256)
void chamfer_init(unsigned* __restrict__ mins, unsigned* __restrict__ maxVal, int n)
{
    int i = blockIdx.x * blockDim.x + threadIdx.x;
    if (i < n) mins[i] = 0x7f800000u;            // +inf bit pattern
    if (i == 0) *maxVal = 0u;
}

// ---------------------------------------------------------------------------
// Main pass. One wave owns (b, 16-row source tile, 1/8th of the targets).
// Per 16x16 tile: ONE v_wmma_f32_16x16x4_f32:
//   D = A(16x4) x B(4x16) + C
//   A row i = (sx, sy, sz, |s|^2)      (K = v + 2h per A-matrix layout)
//   B col j = (-2tx, -2ty, -2tz, 1)    (K = v + 2h per B-matrix layout)
//   C[i,j]  = |t_j|^2                  (per-lane broadcast, column = lane&15)
// D element (r, lane=(l,h)) = distance(source nt*16 + r + 8h, target mt*16+l).
//
// Reductions (nonneg floats == their uint ordering, so u32 atomics are exact):
//   per-lane min over 8 rows -> combine halves (shfl_xor 16) -> dstMin[b,m]
//   per-row running min over this lane's column stream, split by target mask
//     -> cross-lane min at end -> srcMinM/srcMinU[b,n] via atomic u32-min
//   running max -> atomic u32-max (global max_dist)
// ---------------------------------------------------------------------------
__global__ __launch_bounds__(256)
void chamfer_pass(const float* __restrict__ src, const float* __restrict__ tgt,
                  const float* __restrict__ mask,
                  unsigned* __restrict__ srcMinM, unsigned* __restrict__ srcMinU,
                  unsigned* __restrict__ dstMin, unsigned* __restrict__ maxVal)
{
    const int lane = threadIdx.x & 31;
    const int wave = (blockIdx.x * blockDim.x + threadIdx.x) >> 5;  // 0..B_*NT*MCH-1
    const int mc   = wave % MCH;
    const int bt   = wave / MCH;
    const int nt   = bt % NT;
    const int b    = bt / NT;
    const int l    = lane & 15;     // column index / row-in-tile
    const int h    = lane >> 4;     // half select

    // ---- A tile: this wave's 16 source rows --------------------------------
    const int    srow = nt * 16 + l;
    const float* sp   = src + ((size_t)b * N_ + srow) * 3;
    const float  sx = sp[0], sy = sp[1], sz = sp[2];
    const float  sn = sx * sx + sy * sy + sz * sz;

    v2f A;
    A.x = h ? sz : sx;   // K=2 : K=0
    A.y = h ? sn : sy;   // K=3 : K=1

    float rowMinM[8], rowMinU[8];
#pragma unroll
    for (int r = 0; r < 8; ++r) { rowMinM[r] = finf(); rowMinU[r] = finf(); }
    float maxAcc = 0.0f;

    const int mt0 = mc * MTC;
    for (int mt = mt0; mt < mt0 + MTC; ++mt) {
        const int    trow = mt * 16 + l;
        const float* tp   = tgt + ((size_t)b * M_ + trow) * 3;
        const float  tx = tp[0], ty = tp[1], tz = tp[2];
        const float  tn = tx * tx + ty * ty + tz * tz;
        const float  mv = mask[(size_t)b * M_ + trow];   // this column's mask

        v2f Bv;
        Bv.x = h ? (-2.0f * tz) : (-2.0f * tx);  // K=2 : K=0
        Bv.y = h ? 1.0f         : (-2.0f * ty);  // K=3 : K=1

        v8f C;
#pragma unroll
        for (int r = 0; r < 8; ++r) C[r] = tn;   // ||t_col||^2 for every row

        v8f D = __builtin_amdgcn_wmma_f32_16x16x4_f32(
            false, A, false, Bv, (short)0, C, false, false);

        const bool masked = mv > 0.5f;
        float colmin = finf();
#pragma unroll
        for (int r = 0; r < 8; ++r) {
            const float d = D[r];
            maxAcc = fmaxf(maxAcc, d);
            colmin = fminf(colmin, d);
            rowMinM[r] = fminf(rowMinM[r], masked ? d : finf());
            rowMinU[r] = fminf(rowMinU[r], masked ? finf() : d);
        }
        // combine row-halves (M 0-7 with M 8-15) for this column's min
        colmin = fminf(colmin, __shfl_xor(colmin, 16));
        if (h == 0)
            atomicMin(&dstMin[(size_t)b * M_ + trow], __float_as_uint(colmin));
    }

    // ---- row mins: reduce across the 16 lanes of each half -----------------
#pragma unroll
    for (int s = 1; s <= 8; s <<= 1) {
#pragma unroll
        for (int r = 0; r < 8; ++r) {
            rowMinM[r] = fminf(rowMinM[r], __shfl_xor(rowMinM[r], s));
            rowMinU[r] = fminf(rowMinU[r], __shfl_xor(rowMinU[r], s));
        }
    }
    if (l == 0) {  // lanes 0 and 16 own rows 0-7 / 8-15 of this source tile
        const size_t base = (size_t)b * N_ + nt * 16 + 8 * h;
#pragma unroll
        for (int r = 0; r < 8; ++r) {
            atomicMin(&srcMinM[base + r], __float_as_uint(rowMinM[r]));
            atomicMin(&srcMinU[base + r], __float_as_uint(rowMinU[r]));
        }
    }

    // ---- global max --------------------------------------------------------
#pragma unroll
    for (int s = 1; s <= 16; s <<= 1)
        maxAcc = fmaxf(maxAcc, __shfl_xor(maxAcc, s));
    if (lane == 0)
        atomicMax(maxVal, __float_as_uint(maxAcc));
}

// ---------------------------------------------------------------------------
// Finalize: loss_src = mean( min(minMasked - max, minUnmasked) + max )
//           loss_dst = sum(dstMin * mask) / sum(mask)
// (min_m(d - mask*max) + max == min(minMasked - max, minUnmasked) + max
//  exactly in FP: subtracting a common constant is monotone.)
// ---------------------------------------------------------------------------
__global__ __launch_bounds__(256)
void chamfer_finalize(const unsigned* __restrict__ srcMinM,
                      const unsigned* __restrict__ srcMinU,
                      const unsigned* __restrict__ dstMin,
                      const float* __restrict__ mask,
                      const unsigned* __restrict__ maxVal,
                      float* __restrict__ out)
{
    const int tid = threadIdx.x;
    const float maxd = __uint_as_float(*maxVal);

    float sSrc = 0.0f;
    for (int i = tid; i < B_ * N_; i += 256) {
        const float mM = __uint_as_float(srcMinM[i]);
        const float mU = __uint_as_float(srcMinU[i]);
        sSrc += fminf(mM - maxd, mU) + maxd;
    }
    float sDst = 0.0f, sMask = 0.0f;
    for (int i = tid; i < B_ * M_; i += 256) {
        const float mv = mask[i];
        sDst  += __uint_as_float(dstMin[i]) * mv;
        sMask += mv;
    }

    __shared__ float sh[3 * 256];
    sh[tid] = sSrc; sh[256 + tid] = sDst; sh[512 + tid] = sMask;
    __syncthreads();
    for (int s = 128; s > 0; s >>= 1) {
        if (tid < s) {
            sh[tid]       += sh[tid + s];
            sh[256 + tid] += sh[256 + tid + s];
            sh[512 + tid] += sh[512 + tid + s];
        }
        __syncthreads();
    }
    if (tid == 0) {
        out[0] = sh[0] / (float)(B_ * N_);
        out[1] = sh[256] / sh[512];
    }
}

// ---------------------------------------------------------------------------
extern "C" void kernel_launch(void* const* d_in, const int* in_sizes, int n_in,
                              void* d_out, int out_size, void* d_ws, size_t ws_size,
                              hipStream_t stream)
{
    (void)in_sizes; (void)n_in; (void)out_size; (void)ws_size;

    const float* src  = (const float*)d_in[0];   // [B,N,3]
    const float* tgt  = (const float*)d_in[1];   // [B,M,3]
    const float* mask = (const float*)d_in[2];   // [B,M]
    float* out = (float*)d_out;                  // 2 scalars

    // workspace layout (4B elements):
    //   [0,        B*N)        srcMinMasked   (uint)
    //   [B*N,    2*B*N)        srcMinUnmasked (uint)
    //   [2*B*N, 2*B*N+B*M)     dstMin         (uint)
    //   [2*B*N+B*M]            maxVal         (uint)
    unsigned* ws      = (unsigned*)d_ws;
    unsigned* srcMinM = ws;
    unsigned* srcMinU = ws + (size_t)B_ * N_;
    unsigned* dstMin  = ws + 2 * (size_t)B_ * N_;
    unsigned* maxVal  = ws + 2 * (size_t)B_ * N_ + (size_t)B_ * M_;

    const int nInit = 2 * B_ * N_ + B_ * M_;     // all three min arrays
    chamfer_init<<<(nInit + 255) / 256, 256, 0, stream>>>(srcMinM, maxVal, nInit);

    // B_*NT*MCH = 8192 waves; 8 waves per 256-thread block -> 1024 blocks
    chamfer_pass<<<(B_ * NT * MCH) / 8, 256, 0, stream>>>(
        src, tgt, mask, srcMinM, srcMinU, dstMin, maxVal);

    chamfer_finalize<<<1, 256, 0, stream>>>(
        srcMinM, srcMinU, dstMin, mask, maxVal, out);
}